// gru_simp_10187662426390
// MI455X (gfx1250) — compile-verified
//
#include <hip/hip_runtime.h>

// ---------------------------------------------------------------------------
// Fused GRU autoencoder for MI455X (gfx1250, wave32, WGP).
//
// SEQ=1 collapses both GRU scans to a single step. The whole network is
//   enc(250 scalar lanes) -> fc1 GEMV 512x1000 -> fc2 GEMV 250x512 -> dec(250)
// ~1.3 MFLOP / ~2.6 MB: pure latency problem -> ONE kernel, ONE workgroup
// (1024 threads = 32 waves), activations live in LDS, barriers between phases.
// GEMVs use V_WMMA_F32_16X16X4_F32 (full fp32, matches reference precision):
// A = 16-row weight tile per wave, B = activation vector broadcast across all
// 16 columns, so every D column equals the GEMV result for that row tile.
// ---------------------------------------------------------------------------

typedef float v2f __attribute__((ext_vector_type(2)));
typedef float v8f __attribute__((ext_vector_type(8)));

__device__ __forceinline__ float sigf(float t) {
    return 1.0f / (1.0f + __expf(-t));
}

extern "C" __global__ __launch_bounds__(1024)
void gru_autoenc_fused(const float* __restrict__ x,        // (250,)
                       const float* __restrict__ enc_Wih,  // (12,1)
                       const float* __restrict__ enc_bih,  // (12,)
                       const float* __restrict__ enc_bhh,  // (12,)
                       const float* __restrict__ fc1_W,    // (512,1000)
                       const float* __restrict__ fc1_b,    // (512,)
                       const float* __restrict__ fc2_W,    // (250,512)
                       const float* __restrict__ fc2_b,    // (250,)
                       const float* __restrict__ dec_Wih,  // (3,4)
                       const float* __restrict__ dec_Whh,  // (3,1)
                       const float* __restrict__ dec_bih,  // (3,)
                       const float* __restrict__ dec_bhh,  // (3,)
                       float* __restrict__ out)            // (250,)
{
    // hvec = encoder hN flattened b-major (1000) == fc1 input == decoder y.
    __shared__ float hvec[1000];
    __shared__ float h512[512];
    __shared__ float hn[256];

    const int tid  = threadIdx.x;
    const int lane = tid & 31;
    const int wave = tid >> 5;
    const int m    = lane & 15;          // WMMA A-matrix row within tile
    const int half = lane >> 4;          // 0: K=0,1   1: K=2,3
    const int koff = half << 1;

    // ---------------- Phase 1: encoder GRU step (h0 = 0) -------------------
    if (tid < 250) {
        const float xb = x[tid];
        #pragma unroll
        for (int j = 0; j < 4; ++j) {
            const float ir  = fmaf(xb, enc_Wih[j],     enc_bih[j])     + enc_bhh[j];
            const float iz  = fmaf(xb, enc_Wih[4 + j], enc_bih[4 + j]) + enc_bhh[4 + j];
            const float inn = fmaf(xb, enc_Wih[8 + j], enc_bih[8 + j]);
            const float r = sigf(ir);
            const float z = sigf(iz);
            const float n = tanhf(fmaf(r, enc_bhh[8 + j], inn));
            hvec[tid * 4 + j] = (1.0f - z) * n;        // h_new = (1-z)*n + z*0
        }
    }
    __syncthreads();

    // ---------------- Phase 2: fc1 = relu(W(512x1000) @ hvec + b) ----------
    {
        const int rowbase = wave * 16;                 // 32 waves x 16 rows = 512
        const float* ap = fc1_W + (rowbase + m) * 1000 + koff;
        v8f c = {0.f, 0.f, 0.f, 0.f, 0.f, 0.f, 0.f, 0.f};
        for (int kb = 0; kb < 1000; kb += 4) {
            v2f a = *(const v2f*)(ap + kb);                     // W[row][kb+koff .. +1]
            v2f b = *(const v2f*)(&hvec[kb + koff]);           // v broadcast into all N
            c = __builtin_amdgcn_wmma_f32_16x16x4_f32(false, a, false, b,
                                                      (short)0, c, false, false);
        }
        if (m == 0) {                                  // lane 0: rows +0..7, lane 16: +8..15
            const int base = rowbase + (half << 3);
            #pragma unroll
            for (int g = 0; g < 8; ++g) {
                const float v = c[g] + fc1_b[base + g];
                h512[base + g] = v > 0.f ? v : 0.f;
            }
        }
    }
    __syncthreads();

    // ---------------- Phase 3: fc2 = W(250x512) @ h512 + b -----------------
    if (wave < 16) {                                   // wave-uniform branch: EXEC stays full
        const int rowbase = wave * 16;                 // rows 0..255 (250..255 discarded)
        const int row     = rowbase + m;
        const int rowc    = row < 249 ? row : 249;     // clamp OOB loads (outputs discarded)
        const float* ap = fc2_W + rowc * 512 + koff;
        v8f c = {0.f, 0.f, 0.f, 0.f, 0.f, 0.f, 0.f, 0.f};
        for (int kb = 0; kb < 512; kb += 4) {
            v2f a = *(const v2f*)(ap + kb);
            v2f b = *(const v2f*)(&h512[kb + koff]);
            c = __builtin_amdgcn_wmma_f32_16x16x4_f32(false, a, false, b,
                                                      (short)0, c, false, false);
        }
        if (m == 0) {
            const int base = rowbase + (half << 3);
            #pragma unroll
            for (int g = 0; g < 8; ++g) {
                const int r = base + g;
                if (r < 250) hn[r] = c[g] + fc2_b[r];
            }
        }
    }
    __syncthreads();

    // ---------------- Phase 4: decoder GRU step ----------------------------
    if (tid < 250) {
        const float h  = hn[tid];
        const float x0 = hvec[tid * 4 + 0];
        const float x1 = hvec[tid * 4 + 1];
        const float x2 = hvec[tid * 4 + 2];
        const float x3 = hvec[tid * 4 + 3];
        float gi[3];
        #pragma unroll
        for (int k = 0; k < 3; ++k) {
            gi[k] = dec_bih[k]
                  + dec_Wih[k * 4 + 0] * x0 + dec_Wih[k * 4 + 1] * x1
                  + dec_Wih[k * 4 + 2] * x2 + dec_Wih[k * 4 + 3] * x3;
        }
        const float gh0 = fmaf(dec_Whh[0], h, dec_bhh[0]);
        const float gh1 = fmaf(dec_Whh[1], h, dec_bhh[1]);
        const float gh2 = fmaf(dec_Whh[2], h, dec_bhh[2]);
        const float r = sigf(gi[0] + gh0);
        const float z = sigf(gi[1] + gh1);
        const float n = tanhf(fmaf(r, gh2, gi[2]));
        out[tid] = (1.0f - z) * n + z * h;
    }
}

extern "C" void kernel_launch(void* const* d_in, const int* in_sizes, int n_in,
                              void* d_out, int out_size, void* d_ws, size_t ws_size,
                              hipStream_t stream) {
    (void)in_sizes; (void)n_in; (void)out_size; (void)d_ws; (void)ws_size;
    const float* x       = (const float*)d_in[0];
    const float* enc_Wih = (const float*)d_in[1];
    // d_in[2] = enc_Whh: unused (h0 == 0 -> gh reduces to bhh)
    const float* enc_bih = (const float*)d_in[3];
    const float* enc_bhh = (const float*)d_in[4];
    const float* fc1_W   = (const float*)d_in[5];
    const float* fc1_b   = (const float*)d_in[6];
    const float* fc2_W   = (const float*)d_in[7];
    const float* fc2_b   = (const float*)d_in[8];
    const float* dec_Wih = (const float*)d_in[9];
    const float* dec_Whh = (const float*)d_in[10];
    const float* dec_bih = (const float*)d_in[11];
    const float* dec_bhh = (const float*)d_in[12];

    gru_autoenc_fused<<<dim3(1), dim3(1024), 0, stream>>>(
        x, enc_Wih, enc_bih, enc_bhh, fc1_W, fc1_b, fc2_W, fc2_b,
        dec_Wih, dec_Whh, dec_bih, dec_bhh, (float*)d_out);
}